// AdaptiveFNO2d_73065983640193
// MI455X (gfx1250) — compile-verified
//
#include <hip/hip_runtime.h>
#include <math.h>

// ---------------------------------------------------------------------------
// FNO2d forward for MI455X (gfx1250, wave32).
// All dense math (DFTs-as-matmuls, spectral einsum, 1x1 conv) runs on
// V_WMMA_F32_16X16X4_F32.  Spectral ky-dim padded 65 -> 80 (zero columns) so
// every WMMA tile is full.  Hot B-operands staged through LDS; wr stream
// prefetched with global_prefetch_b8.
// ---------------------------------------------------------------------------

typedef float v2f __attribute__((ext_vector_type(2)));
typedef float v8f __attribute__((ext_vector_type(8)));

#define PI_F 3.14159265358979323846f

static constexpr int BATCH = 16, UD = 3, NX = 128, NY = 128, OY = 65, OYP = 80;
static constexpr int WID = 32, NLAY = 4;
static constexpr int PLANE = NX * NY;                       // 16384
static constexpr int ROWS  = BATCH * WID * NX;              // 65536 (b,c,x) rows
static constexpr int NMODE = NX * OY;                       // 8320

// ---- workspace layout (float offsets) ----
static constexpr size_t SZ_X  = (size_t)BATCH * WID * PLANE;   // 8388608
static constexpr size_t SZ_F  = (size_t)ROWS * OYP;            // 5242880
static constexpr size_t OFF_X0     = 0;
static constexpr size_t OFF_X1     = OFF_X0 + SZ_X;
static constexpr size_t OFF_FYRE   = OFF_X1 + SZ_X;
static constexpr size_t OFF_FYIM   = OFF_FYRE + SZ_F;
static constexpr size_t OFF_FXRE   = OFF_FYIM + SZ_F;
static constexpr size_t OFF_FXIM   = OFF_FXRE + SZ_F;
static constexpr size_t OFF_WYF_RE = OFF_FXIM + SZ_F;          // [128][80]
static constexpr size_t OFF_WYF_IM = OFF_WYF_RE + 128 * OYP;
static constexpr size_t OFF_WXF_RE = OFF_WYF_IM + 128 * OYP;   // [kx][x]
static constexpr size_t OFF_WXF_IM = OFF_WXF_RE + 128 * 128;
static constexpr size_t OFF_WXI_RE = OFF_WXF_IM + 128 * 128;   // [x'][kx]
static constexpr size_t OFF_WXI_IM = OFF_WXI_RE + 128 * 128;
static constexpr size_t OFF_WYI_C  = OFF_WXI_IM + 128 * 128;   // [80][128]
static constexpr size_t OFF_WYI_S  = OFF_WYI_C + OYP * 128;
static constexpr size_t OFF_S      = OFF_WYI_S + OYP * 128;    // [128*65]
static constexpr size_t OFF_MODES  = OFF_S + NMODE;            // 2 ints
static constexpr size_t WS_FLOATS  = OFF_MODES + 2;

__device__ __forceinline__ v8f wmma4(v2f a, v2f b, v8f c) {
  // D = A(16x4 f32) * B(4x16 f32) + C(16x16 f32)
  return __builtin_amdgcn_wmma_f32_16x16x4_f32(false, a, false, b,
                                               (short)0, c, false, false);
}

__device__ __forceinline__ float gelu_exact(float v) {
  return 0.5f * v * (1.0f + erff(v * 0.70710678118654752440f));
}

// ---------------------------------------------------------------------------
// Twiddle tables (padded with zeros in the ky direction).
// ---------------------------------------------------------------------------
__global__ void k_tables(float* __restrict__ wyfre, float* __restrict__ wyfim,
                         float* __restrict__ wxfre, float* __restrict__ wxfim,
                         float* __restrict__ wxire, float* __restrict__ wxiim,
                         float* __restrict__ wyic,  float* __restrict__ wyis) {
  int idx = blockIdx.x * 256 + threadIdx.x;
  if (idx < 128 * OYP) {                       // forward DFT along Y: [y][ky]
    int y = idx / OYP, ky = idx % OYP;
    float re = 0.f, im = 0.f;
    if (ky < OY) {
      float a = 2.f * PI_F * (float)(y * ky) * (1.f / 128.f);
      re = cosf(a); im = -sinf(a);
    }
    wyfre[idx] = re; wyfim[idx] = im;
  }
  if (idx < 128 * 128) {                       // X DFTs: symmetric index
    int r = idx / 128, c = idx % 128;
    float a = 2.f * PI_F * (float)(r * c) * (1.f / 128.f);
    wxfre[idx] = cosf(a);                 wxfim[idx] = -sinf(a);
    wxire[idx] = cosf(a) * (1.f / 128.f); wxiim[idx] = sinf(a) * (1.f / 128.f);
  }
  if (idx < OYP * 128) {                       // inverse along Y: [ky][y]
    int ky = idx / 128, y = idx % 128;
    float cv = 0.f, sv = 0.f;
    if (ky < OY) {
      float w = (ky == 0 || ky == OY - 1) ? 1.f : 2.f;
      float a = 2.f * PI_F * (float)(ky * y) * (1.f / 128.f);
      cv = w * cosf(a) * (1.f / 128.f);
      sv = w * sinf(a) * (1.f / 128.f);
    }
    wyic[idx] = cv; wyis[idx] = sv;
  }
}

// ---------------------------------------------------------------------------
// Lift: x0[b,w,p] = P_b[w] + sum_u P_w[w,u] * in[b,u,p]
// ---------------------------------------------------------------------------
__global__ void k_lift(const float* __restrict__ in, const float* __restrict__ Pw,
                       const float* __restrict__ Pb, float* __restrict__ x0) {
  size_t tid = (size_t)blockIdx.x * 256 + threadIdx.x;
  if (tid >= SZ_X) return;
  int p = (int)(tid % PLANE);
  int w = (int)((tid / PLANE) % WID);
  int b = (int)(tid / ((size_t)PLANE * WID));
  const float* ib = in + (size_t)b * UD * PLANE + p;
  float acc = Pb[w];
  #pragma unroll
  for (int u = 0; u < UD; ++u) acc += Pw[w * UD + u] * ib[(size_t)u * PLANE];
  x0[tid] = acc;
}

// ---------------------------------------------------------------------------
// Mode energy: s[m] = sqrt( sum_{io} |wr[layer,io,m]|^2 )   (block per mode)
// ---------------------------------------------------------------------------
__global__ void k_modes_s(const float2* __restrict__ wr, int layer,
                          float* __restrict__ s) {
  __shared__ float red[256];
  int m = blockIdx.x;
  size_t base = (size_t)layer * 1024 * NMODE + m;
  float acc = 0.f;
  for (int j = threadIdx.x; j < 1024; j += 256) {
    float2 v = wr[base + (size_t)j * NMODE];
    acc += v.x * v.x + v.y * v.y;
  }
  red[threadIdx.x] = acc;
  __syncthreads();
  for (int st = 128; st > 0; st >>= 1) {
    if (threadIdx.x < st) red[threadIdx.x] += red[threadIdx.x + st];
    __syncthreads();
  }
  if (threadIdx.x == 0) s[m] = sqrtf(red[0]);
}

// Row-major 2D cumsum scan; first (i,j) with ratio >= 0.99 -> keep [:i,:j]
__global__ void k_modes_sel(const float* __restrict__ s, int* __restrict__ modes) {
  __shared__ float red[128];
  __shared__ float colpref[OY];
  __shared__ float thresh_s;
  __shared__ int found, mi, mj;
  float acc = 0.f;
  for (int i = threadIdx.x; i < NMODE; i += 128) acc += s[i];
  red[threadIdx.x] = acc;
  __syncthreads();
  for (int st = 64; st > 0; st >>= 1) {
    if (threadIdx.x < st) red[threadIdx.x] += red[threadIdx.x + st];
    __syncthreads();
  }
  if (threadIdx.x == 0) { thresh_s = 0.99f * red[0]; found = 0; mi = 0; mj = 0; }
  if (threadIdx.x < OY) colpref[threadIdx.x] = 0.f;
  __syncthreads();
  for (int i = 0; i < NX; ++i) {
    if (threadIdx.x < OY) colpref[threadIdx.x] += s[i * OY + threadIdx.x];
    __syncthreads();
    if (threadIdx.x == 0 && !found) {
      float run = 0.f;
      for (int j = 0; j < OY; ++j) {
        run += colpref[j];
        if (run >= thresh_s) { found = 1; mi = i; mj = j; break; }
      }
    }
    __syncthreads();
  }
  if (threadIdx.x == 0) { modes[0] = mi; modes[1] = mj; }
}

// ---------------------------------------------------------------------------
// Forward DFT along Y (real -> complex): rows (b,c,x) x 128  ->  x OYP.
// One wave per 16x16 tile; 2 WMMAs per k-step; every operand loaded once.
// ---------------------------------------------------------------------------
__global__ void k_dft_y_fwd(const float* __restrict__ x,
                            const float* __restrict__ wre,
                            const float* __restrict__ wim,
                            float* __restrict__ fre, float* __restrict__ fim) {
  int id = blockIdx.x;                       // (ROWS/16) * 5
  int mt = id / 5, nt = id % 5;
  int lane = threadIdx.x;
  int kh = lane >> 4, l15 = lane & 15;
  int mrow = mt * 16 + l15;
  int n = nt * 16 + l15;
  const v2f* arow = (const v2f*)(x + (size_t)mrow * NY);
  v8f cre = {0.f, 0.f, 0.f, 0.f, 0.f, 0.f, 0.f, 0.f};
  v8f cim = cre;
  for (int kk = 0; kk < NY / 4; ++kk) {
    int k0 = kk * 4 + kh * 2;
    v2f a = arow[k0 >> 1];
    v2f bre, bim;
    bre.x = wre[k0 * OYP + n];       bre.y = wre[(k0 + 1) * OYP + n];
    bim.x = wim[k0 * OYP + n];       bim.y = wim[(k0 + 1) * OYP + n];
    cre = wmma4(a, bre, cre);
    cim = wmma4(a, bim, cim);
  }
  #pragma unroll
  for (int r = 0; r < 8; ++r) {
    size_t row = (size_t)(mt * 16 + r + 8 * kh);
    fre[row * OYP + n] = cre[r];
    fim[row * OYP + n] = cim[r];
  }
}

// ---------------------------------------------------------------------------
// Complex DFT along X (forward/inverse via tables).  Block = 8 waves covering
// all kx tiles of one (bc, nt); the 128x16 spectrum slice is staged in LDS
// once (8x traffic reduction) and feeds the WMMA B operands via ds_load.
// ---------------------------------------------------------------------------
__global__ void k_dft_x(const float* __restrict__ fre_in,
                        const float* __restrict__ fim_in,
                        const float* __restrict__ tre,
                        const float* __restrict__ tim,
                        float* __restrict__ fre_out,
                        float* __restrict__ fim_out) {
  __shared__ float sRe[128 * 16];
  __shared__ float sIm[128 * 16];
  int id = blockIdx.x;                       // 512 * 5
  int bc = id / 5, nt = id % 5;
  size_t base = (size_t)bc * NX * OYP;
  int nBase = nt * 16;
  // cooperative stage of the B slice [k=0..127][n=0..15]
  for (int t = threadIdx.x; t < 128 * 16; t += 256) {
    int k = t >> 4, n = t & 15;
    size_t g = base + (size_t)k * OYP + nBase + n;
    sRe[t] = fre_in[g];
    sIm[t] = fim_in[g];
  }
  __syncthreads();

  int mt = threadIdx.x >> 5;                 // wave id = kx tile
  int lane = threadIdx.x & 31;
  int kh = lane >> 4, l15 = lane & 15;
  int mrow = mt * 16 + l15;
  const v2f* trr = (const v2f*)(tre + (size_t)mrow * NX);
  const v2f* tri = (const v2f*)(tim + (size_t)mrow * NX);
  v8f cre = {0.f, 0.f, 0.f, 0.f, 0.f, 0.f, 0.f, 0.f};
  v8f cim = cre;
  for (int kk = 0; kk < NX / 4; ++kk) {
    int k0 = kk * 4 + kh * 2;
    v2f are = trr[k0 >> 1];
    v2f aim = tri[k0 >> 1];
    v2f aimn = -aim;
    v2f bre, bim;
    bre.x = sRe[k0 * 16 + l15];       bre.y = sRe[(k0 + 1) * 16 + l15];
    bim.x = sIm[k0 * 16 + l15];       bim.y = sIm[(k0 + 1) * 16 + l15];
    cre = wmma4(are,  bre, cre);
    cre = wmma4(aimn, bim, cre);
    cim = wmma4(are,  bim, cim);
    cim = wmma4(aim,  bre, cim);
  }
  #pragma unroll
  for (int r = 0; r < 8; ++r) {
    size_t idx = base + (size_t)(mt * 16 + r + 8 * kh) * OYP + nBase + l15;
    fre_out[idx] = cre[r];
    fim_out[idx] = cim[r];
  }
}

// ---------------------------------------------------------------------------
// Per-mode spectral einsum, in place:  G[b,o] = sum_i F[b,i] * W[i,o]  (cplx).
// Block = 2 waves (o-tiles).  F block staged once in LDS (bank-padded);
// wr stream (the dominant 272 MB) prefetched one k-step ahead.
// ---------------------------------------------------------------------------
__global__ void k_einsum(float* __restrict__ fre, float* __restrict__ fim,
                         const float2* __restrict__ wr,
                         const int* __restrict__ modes, int layer) {
  __shared__ float sFre[16 * 33];            // [b][i], stride 33: no conflicts
  __shared__ float sFim[16 * 33];
  int m = blockIdx.x;                        // NMODE blocks
  int mx = m / OY, my = m % OY;
  int Mi = modes[0], Mj = modes[1];
  bool kept = (mx < Mi) && (my < Mj);        // block-uniform
  int wave = threadIdx.x >> 5;
  int lane = threadIdx.x & 31;
  int kh = lane >> 4, l15 = lane & 15;
  int ob = wave * 16;
  v8f cre = {0.f, 0.f, 0.f, 0.f, 0.f, 0.f, 0.f, 0.f};
  v8f cim = cre;
  if (kept) {                                // stage F[b=0..15][i=0..31]
    for (int t = threadIdx.x; t < 512; t += 64) {
      int b = t >> 5, i = t & 31;
      size_t idx = ((size_t)(b * WID + i) * NX + mx) * OYP + my;
      sFre[b * 33 + i] = fre[idx];
      sFim[b * 33 + i] = fim[idx];
    }
  }
  __syncthreads();                           // all global reads done here
  if (kept) {
    int b = l15;
    int o = ob + l15;
    size_t wb = (size_t)layer * 1024 * NMODE + m;
    for (int kk = 0; kk < 8; ++kk) {
      int i0 = kk * 4 + kh * 2;
      v2f are, aim;
      are.x = sFre[b * 33 + i0]; are.y = sFre[b * 33 + i0 + 1];
      aim.x = sFim[b * 33 + i0]; aim.y = sFim[b * 33 + i0 + 1];
      v2f aimn = -aim;
      float2 w0 = wr[wb + (size_t)(i0 * WID + o) * NMODE];
      float2 w1 = wr[wb + (size_t)((i0 + 1) * WID + o) * NMODE];
      if (kk < 7)                            // global_prefetch_b8 on wr stream
        __builtin_prefetch(&wr[wb + (size_t)((i0 + 4) * WID + o) * NMODE], 0, 3);
      v2f bre, bim;
      bre.x = w0.x; bre.y = w1.x;
      bim.x = w0.y; bim.y = w1.y;
      cre = wmma4(are,  bre, cre);
      cre = wmma4(aimn, bim, cre);
      cim = wmma4(are,  bim, cim);
      cim = wmma4(aim,  bre, cim);
    }
  }
  #pragma unroll
  for (int r = 0; r < 8; ++r) {
    int b = r + 8 * kh;
    int o = ob + l15;
    size_t idx = ((size_t)(b * WID + o) * NX + mx) * OYP + my;
    fre[idx] = cre[r];
    fim[idx] = cim[r];
  }
}

// ---------------------------------------------------------------------------
// Inverse DFT along Y (complex -> real), hermitian-weighted tables.
// ---------------------------------------------------------------------------
__global__ void k_dft_y_inv(const float* __restrict__ fre,
                            const float* __restrict__ fim,
                            const float* __restrict__ tc,
                            const float* __restrict__ ts,
                            float* __restrict__ xout) {
  int id = blockIdx.x;                       // (ROWS/16) * 8
  int mt = id >> 3, nt = id & 7;
  int lane = threadIdx.x;
  int kh = lane >> 4, l15 = lane & 15;
  int mrow = mt * 16 + l15;
  int n = nt * 16 + l15;
  const v2f* arr = (const v2f*)(fre + (size_t)mrow * OYP);
  const v2f* ari = (const v2f*)(fim + (size_t)mrow * OYP);
  v8f c = {0.f, 0.f, 0.f, 0.f, 0.f, 0.f, 0.f, 0.f};
  for (int kk = 0; kk < OYP / 4; ++kk) {
    int k0 = kk * 4 + kh * 2;
    v2f are = arr[k0 >> 1];
    v2f aim = ari[k0 >> 1];
    v2f aimn = -aim;
    v2f bc_, bs_;
    bc_.x = tc[k0 * 128 + n]; bc_.y = tc[(k0 + 1) * 128 + n];
    bs_.x = ts[k0 * 128 + n]; bs_.y = ts[(k0 + 1) * 128 + n];
    c = wmma4(are,  bc_, c);
    c = wmma4(aimn, bs_, c);
  }
  #pragma unroll
  for (int r = 0; r < 8; ++r)
    xout[(size_t)(mt * 16 + r + 8 * kh) * NY + n] = c[r];
}

// ---------------------------------------------------------------------------
// 1x1 conv as WMMA + bias + residual(o1) + exact GELU.
// Rows = (b, p) tiles of 16, K = 32 in-channels, N = 32 out-channels.
// x_next = gelu( o1 + x*wc^T + bc );  o1 already resides in xo.
// ---------------------------------------------------------------------------
__global__ void k_conv_wmma(const float* __restrict__ xin,
                            const float* __restrict__ wc,
                            const float* __restrict__ bcv,
                            float* __restrict__ xo, int layer) {
  int id = blockIdx.x;                       // BATCH * PLANE/16 = 16384
  int lane = threadIdx.x;
  int kh = lane >> 4, l15 = lane & 15;
  int b = id / (PLANE / 16);
  int pbase = (id % (PLANE / 16)) * 16;
  const float* xb  = xin + (size_t)b * WID * PLANE + pbase + l15;  // A row
  const float* wct = wc + (size_t)layer * WID * WID;
  v8f c0 = {0.f, 0.f, 0.f, 0.f, 0.f, 0.f, 0.f, 0.f};
  v8f c1 = c0;
  for (int kk = 0; kk < 8; ++kk) {
    int k0 = kk * 4 + kh * 2;
    v2f a;
    a.x = xb[(size_t)k0 * PLANE];
    a.y = xb[(size_t)(k0 + 1) * PLANE];
    v2f b0 = *(const v2f*)(wct + (size_t)l15 * WID + k0);         // B[k][o]
    v2f b1 = *(const v2f*)(wct + (size_t)(16 + l15) * WID + k0);
    c0 = wmma4(a, b0, c0);
    c1 = wmma4(a, b1, c1);
  }
  float bias0 = bcv[layer * WID + l15];
  float bias1 = bcv[layer * WID + 16 + l15];
  #pragma unroll
  for (int r = 0; r < 8; ++r) {
    int p = pbase + r + 8 * kh;
    size_t i0 = ((size_t)b * WID + l15) * PLANE + p;
    size_t i1 = ((size_t)b * WID + 16 + l15) * PLANE + p;
    xo[i0] = gelu_exact(xo[i0] + c0[r] + bias0);
    xo[i1] = gelu_exact(xo[i1] + c1[r] + bias1);
  }
}

// ---------------------------------------------------------------------------
// Projection 32 -> 3 + gelu
// ---------------------------------------------------------------------------
__global__ void k_project(const float* __restrict__ x,
                          const float* __restrict__ Qw,
                          const float* __restrict__ Qb,
                          float* __restrict__ out) {
  size_t tid = (size_t)blockIdx.x * 256 + threadIdx.x;
  size_t total = (size_t)BATCH * UD * PLANE;
  if (tid >= total) return;
  int p = (int)(tid % PLANE);
  int u = (int)((tid / PLANE) % UD);
  int b = (int)(tid / ((size_t)PLANE * UD));
  const float* xb = x + (size_t)b * WID * PLANE + p;
  float acc = Qb[u];
  for (int w = 0; w < WID; ++w) acc += Qw[u * WID + w] * xb[(size_t)w * PLANE];
  out[tid] = gelu_exact(acc);
}

// ---------------------------------------------------------------------------
extern "C" void kernel_launch(void* const* d_in, const int* in_sizes, int n_in,
                              void* d_out, int out_size, void* d_ws, size_t ws_size,
                              hipStream_t stream) {
  (void)in_sizes; (void)n_in; (void)out_size;
  if (ws_size < WS_FLOATS * sizeof(float)) return;   // workspace too small

  const float*  in = (const float*)d_in[0];
  const float*  Pw = (const float*)d_in[1];
  const float*  Pb = (const float*)d_in[2];
  const float*  Qw = (const float*)d_in[3];
  const float*  Qb = (const float*)d_in[4];
  const float2* wr = (const float2*)d_in[5];
  const float*  wc = (const float*)d_in[6];
  const float*  bc = (const float*)d_in[7];
  float* wsf = (float*)d_ws;
  int*   modes = (int*)(wsf + OFF_MODES);

  // twiddle tables
  k_tables<<<(128 * 128 + 255) / 256, 256, 0, stream>>>(
      wsf + OFF_WYF_RE, wsf + OFF_WYF_IM, wsf + OFF_WXF_RE, wsf + OFF_WXF_IM,
      wsf + OFF_WXI_RE, wsf + OFF_WXI_IM, wsf + OFF_WYI_C,  wsf + OFF_WYI_S);

  // lift
  k_lift<<<(unsigned)((SZ_X + 255) / 256), 256, 0, stream>>>(in, Pw, Pb, wsf + OFF_X0);

  float* xc = wsf + OFF_X0;
  float* xn = wsf + OFF_X1;
  for (int k = 0; k < NLAY; ++k) {
    k_modes_s<<<NMODE, 256, 0, stream>>>(wr, k, wsf + OFF_S);
    k_modes_sel<<<1, 128, 0, stream>>>(wsf + OFF_S, modes);

    k_dft_y_fwd<<<(ROWS / 16) * 5, 32, 0, stream>>>(
        xc, wsf + OFF_WYF_RE, wsf + OFF_WYF_IM, wsf + OFF_FYRE, wsf + OFF_FYIM);
    k_dft_x<<<BATCH * WID * 5, 256, 0, stream>>>(
        wsf + OFF_FYRE, wsf + OFF_FYIM, wsf + OFF_WXF_RE, wsf + OFF_WXF_IM,
        wsf + OFF_FXRE, wsf + OFF_FXIM);
    k_einsum<<<NMODE, 64, 0, stream>>>(wsf + OFF_FXRE, wsf + OFF_FXIM, wr, modes, k);
    k_dft_x<<<BATCH * WID * 5, 256, 0, stream>>>(
        wsf + OFF_FXRE, wsf + OFF_FXIM, wsf + OFF_WXI_RE, wsf + OFF_WXI_IM,
        wsf + OFF_FYRE, wsf + OFF_FYIM);
    k_dft_y_inv<<<(ROWS / 16) * 8, 32, 0, stream>>>(
        wsf + OFF_FYRE, wsf + OFF_FYIM, wsf + OFF_WYI_C, wsf + OFF_WYI_S, xn);

    k_conv_wmma<<<BATCH * (PLANE / 16), 32, 0, stream>>>(xc, wc, bc, xn, k);

    float* tmp = xc; xc = xn; xn = tmp;
  }

  k_project<<<(unsigned)(((size_t)BATCH * UD * PLANE + 255) / 256), 256, 0, stream>>>(
      xc, Qw, Qb, (float*)d_out);
}